// SAGE_37589553775131
// MI455X (gfx1250) — compile-verified
//
#include <hip/hip_runtime.h>
#include <math.h>

// ---------------------------------------------------------------------------
// GraphSAGE 2-layer forward for MI455X (gfx1250, wave32).
//   * GEMMs: V_WMMA_F32_16X16X4_F32 (fp32 matrix pipe; workload is
//     bandwidth/atomic bound so fp32 preserves reference numerics for free).
//   * Weights staged into LDS by the Tensor Data Mover (tensor_load_to_lds),
//     double-buffered 32-row K-panels, synced with s_wait_tensorcnt +
//     workgroup barriers, prefetch overlapped with WMMA compute.
//   * Scatter (segment_sum) uses hardware global_atomic_add_f32.
// ---------------------------------------------------------------------------

typedef __attribute__((ext_vector_type(2))) float        v2f;
typedef __attribute__((ext_vector_type(8))) float        v8f;
typedef __attribute__((ext_vector_type(4))) unsigned int v4u;
typedef __attribute__((ext_vector_type(8))) int          v8i;
typedef __attribute__((ext_vector_type(4))) int          v4i;

#define SAGE_IN      256
#define SAGE_FEAT    128
#define SAGE_HID     128
#define SAGE_OUT     64

#if __has_builtin(__builtin_amdgcn_tensor_load_to_lds)
#define SAGE_HAS_TDM 1
#else
#define SAGE_HAS_TDM 0
#endif

__device__ __forceinline__ void atomic_add_f32(float* p, float v) {
    unsafeAtomicAdd(p, v);     // lowers to global_atomic_add_f32 on AMD
}

#if SAGE_HAS_TDM
// Issue one TDM load of a [rows x NC] fp32 tile (contiguous rows, row stride
// NC) from gsrc into LDS at byte offset lds_off. Descriptor per CDNA5 D# spec:
//   group0: count=1 | lds_addr | global_addr[56:0] | type=2
//   group1: data_size=4B, tensor_dim0=NC, tensor_dim1=rows,
//           tile_dim0=NC, tile_dim1=rows, tensor_dim0_stride=NC
// This toolchain's builtin takes 6 args (g0, g1, g2, g3, g4, cpol); the 2D
// descriptor only needs groups 0/1, so groups 2-4 are zero.
__device__ __forceinline__ void tdm_load_tile(const float* gsrc,
                                              unsigned lds_off,
                                              int NC, int rows) {
    unsigned long long ga = (unsigned long long)(uintptr_t)gsrc;
    v4u g0 = { 1u,                                   // count = 1 valid descriptor
               lds_off,                              // LDS byte address
               (unsigned)(ga & 0xffffffffu),         // global_addr[31:0]
               (unsigned)(((ga >> 32) & 0x1ffffffu)  // global_addr[56:32]
                          | 0x80000000u) };          // type = 2 (bits 127:126)
    v8i g1 = { (int)(2u << 16),                                  // data_size = 4B
               (int)(((unsigned)NC & 0xffffu) << 16),            // tensor_dim0 lo16
               (int)((((unsigned)NC >> 16) & 0xffffu)
                     | (((unsigned)rows & 0xffffu) << 16)),      // dim0 hi / dim1 lo
               (int)((((unsigned)rows >> 16) & 0xffffu)
                     | (((unsigned)NC & 0xffffu) << 16)),        // dim1 hi / tile_dim0
               (int)((unsigned)rows & 0xffffu),                  // tile_dim1 (tile_dim2=0)
               (int)(unsigned)NC,                                // dim0_stride lo32
               0, 0 };
    v4i z4 = {0, 0, 0, 0};
    v8i z8 = {0, 0, 0, 0, 0, 0, 0, 0};
    __builtin_amdgcn_tensor_load_to_lds(g0, g1, z4, z4, z8, 0);
}
#endif

// --------------------------- utility kernels -------------------------------

__global__ void sage_zero_f32(float* __restrict__ p, int n) {
    int i = blockIdx.x * blockDim.x + threadIdx.x;
    int stride = gridDim.x * blockDim.x;
    for (; i < n; i += stride) p[i] = 0.0f;
}

__global__ void sage_deg_count(const int* __restrict__ dst, float* __restrict__ deg, int E) {
    int e = blockIdx.x * blockDim.x + threadIdx.x;
    if (e < E) atomic_add_f32(&deg[dst[e]], 1.0f);
}

__global__ void sage_deg_inv(float* __restrict__ deg, int n) {
    int i = blockIdx.x * blockDim.x + threadIdx.x;
    if (i < n) {
        float d = deg[i];
        deg[i] = (d > 0.0f) ? (1.0f / d) : 0.0f;
    }
}

// aggr[dst[e], :] += h[src[e], :] -- one thread per 4 features of one edge.
__global__ void sage_scatter_add(const float* __restrict__ h,
                                 const int* __restrict__ src,
                                 const int* __restrict__ dst,
                                 float* __restrict__ aggr,
                                 int E, int D) {
    int idx = blockIdx.x * blockDim.x + threadIdx.x;
    int chunks = D >> 2;
    int total = E * chunks;
    if (idx >= total) return;
    int e = idx / chunks;
    int c = (idx - e * chunks) << 2;
    int s = src[e];
    int d = dst[e];
    const float4 v = *(const float4*)(h + (size_t)s * D + c);
    float* ap = aggr + (size_t)d * D + c;
    atomic_add_f32(ap + 0, v.x);
    atomic_add_f32(ap + 1, v.y);
    atomic_add_f32(ap + 2, v.z);
    atomic_add_f32(ap + 3, v.w);
}

// --------------------------- fused WMMA GEMM -------------------------------
// O = diag(rowscale?) * A @ W  (+ A2 @ W2 if DUAL) + bias, optional ReLU.
// Block = 256 threads = 8 wave32s; each wave owns one 16x16 output tile.
// W / W2 are staged into LDS as double-buffered 32-row K-panels by the TDM
// (wave 0 issues, s_wait_tensorcnt, barrier), prefetch overlapped with WMMA.
template <int K, int NC, bool DUAL>
__global__ __launch_bounds__(256) void sage_gemm_wmma(
    const float* __restrict__ A, const float* __restrict__ W,
    const float* __restrict__ A2, const float* __restrict__ W2,
    const float* __restrict__ bias, const float* __restrict__ rowscale,
    float* __restrict__ O, int mTiles, int doRelu) {

    constexpr int CHUNK       = 32;              // K rows per staged panel
    constexpr int NMAT        = DUAL ? 2 : 1;
    constexpr int CHUNK_ELEMS = CHUNK * NC;      // per matrix
    constexpr int BUF_ELEMS   = CHUNK_ELEMS * NMAT;
    constexpr int NCHUNKS     = K / CHUNK;
    __shared__ float lds_w[2 * BUF_ELEMS];       // <= 64KB in every instantiation

    const int wave = threadIdx.x >> 5;
    const int lane = threadIdx.x & 31;
    constexpr int nTiles         = NC / 16;
    constexpr int stripsPerBlock = 8 / nTiles;
    const int mTile  = blockIdx.x * stripsPerBlock + (wave / nTiles);
    const int nTile  = wave & (nTiles - 1);
    const bool active = (mTile < mTiles);

    const int half  = lane >> 4;       // which K-pair this lane supplies
    const int l16   = lane & 15;
    const int kOff  = half << 1;       // 0 or 2
    const int mBase = mTile << 4;
    const int nBase = nTile << 4;
    const int row   = active ? (mBase + l16) : 0;

    v8f acc = {0.0f, 0.0f, 0.0f, 0.0f, 0.0f, 0.0f, 0.0f, 0.0f};
    const float s = rowscale ? rowscale[row] : 1.0f;
    const float* Ap  = A + (size_t)row * K + kOff;
    const float* A2p = DUAL ? (A2 + (size_t)row * K + kOff) : nullptr;

    auto compute_panel = [&](const float* buf, int kb) {
        const float* Wb = buf;
#pragma unroll
        for (int r = 0; r < CHUNK; r += 4) {
            v2f a, b;
            a.x = Ap[kb + r] * s;
            a.y = Ap[kb + r + 1] * s;
            b.x = Wb[(r + kOff) * NC + nBase + l16];
            b.y = Wb[(r + kOff + 1) * NC + nBase + l16];
            acc = __builtin_amdgcn_wmma_f32_16x16x4_f32(
                false, a, false, b, (short)0, acc, false, false);
        }
        if (DUAL) {
            const float* W2b = buf + CHUNK_ELEMS;
#pragma unroll
            for (int r = 0; r < CHUNK; r += 4) {
                v2f a, b;
                a.x = A2p[kb + r];
                a.y = A2p[kb + r + 1];
                b.x = W2b[(r + kOff) * NC + nBase + l16];
                b.y = W2b[(r + kOff + 1) * NC + nBase + l16];
                acc = __builtin_amdgcn_wmma_f32_16x16x4_f32(
                    false, a, false, b, (short)0, acc, false, false);
            }
        }
    };

#if SAGE_HAS_TDM
    const unsigned lds_base = (unsigned)(uintptr_t)(&lds_w[0]);
    if (wave == 0) {                                  // prime panel 0
        tdm_load_tile(W, lds_base, NC, CHUNK);
        if (DUAL) tdm_load_tile(W2, lds_base + CHUNK_ELEMS * 4, NC, CHUNK);
    }
    for (int c = 0; c < NCHUNKS; ++c) {
        if (wave == 0) {
            if (c + 1 < NCHUNKS) {                    // prefetch next panel
                const unsigned off = lds_base + (unsigned)(((c + 1) & 1) * BUF_ELEMS * 4);
                tdm_load_tile(W + (size_t)(c + 1) * CHUNK_ELEMS, off, NC, CHUNK);
                if (DUAL)
                    tdm_load_tile(W2 + (size_t)(c + 1) * CHUNK_ELEMS,
                                  off + CHUNK_ELEMS * 4, NC, CHUNK);
                __builtin_amdgcn_s_wait_tensorcnt(NMAT);  // panel c done (in-order)
            } else {
                __builtin_amdgcn_s_wait_tensorcnt(0);     // drain
            }
        }
        __syncthreads();                              // panel c visible to all waves
        if (active) compute_panel(&lds_w[(c & 1) * BUF_ELEMS], c * CHUNK);
        __syncthreads();                              // done reading before reuse
    }
#else
    // Fallback: cooperative synchronous staging (no TDM builtin available).
    for (int c = 0; c < NCHUNKS; ++c) {
        for (int i = threadIdx.x; i < CHUNK_ELEMS; i += 256) {
            lds_w[i] = W[(size_t)c * CHUNK_ELEMS + i];
            if (DUAL) lds_w[CHUNK_ELEMS + i] = W2[(size_t)c * CHUNK_ELEMS + i];
        }
        __syncthreads();
        if (active) compute_panel(&lds_w[0], c * CHUNK);
        __syncthreads();
    }
#endif

    if (!active) return;
    const float bv = bias ? bias[nBase + l16] : 0.0f;
    float* Op = O + (size_t)(mBase + half * 8) * NC + nBase + l16;
#pragma unroll
    for (int r = 0; r < 8; ++r) {
        float v = acc[r] + bv;
        if (doRelu) v = fmaxf(v, 0.0f);
        Op[(size_t)r * NC] = v;
    }
}

// --------------------------- row L2 normalize ------------------------------
__global__ void sage_l2norm_rows(float* __restrict__ out, int nrows) {
    const int wave = (blockIdx.x * blockDim.x + threadIdx.x) >> 5;
    const int lane = threadIdx.x & 31;
    if (wave >= nrows) return;
    float* row = out + (size_t)wave * SAGE_OUT;
    float v0 = row[lane];
    float v1 = row[lane + 32];
    float ss = v0 * v0 + v1 * v1;
#pragma unroll
    for (int m = 16; m > 0; m >>= 1) ss += __shfl_xor(ss, m, 32);
    float inv = 1.0f / fmaxf(sqrtf(ss), 1e-12f);
    row[lane]      = v0 * inv;
    row[lane + 32] = v1 * inv;
}

// ------------------------------- launcher ----------------------------------

static inline int cdiv(int a, int b) { return (a + b - 1) / b; }

extern "C" void kernel_launch(void* const* d_in, const int* in_sizes, int n_in,
                              void* d_out, int out_size, void* d_ws, size_t ws_size,
                              hipStream_t stream) {
    const float* x     = (const float*)d_in[0];
    const int*   edge  = (const int*)d_in[1];
    const float* W_pre = (const float*)d_in[2];
    const float* b_pre = (const float*)d_in[3];
    const float* Wl1   = (const float*)d_in[4];
    const float* bl1   = (const float*)d_in[5];
    const float* Wr1   = (const float*)d_in[6];
    const float* Wl2   = (const float*)d_in[7];
    const float* bl2   = (const float*)d_in[8];
    const float* Wr2   = (const float*)d_in[9];

    const int N = in_sizes[0] / SAGE_IN;     // 50000
    const int E = in_sizes[1] / 2;           // 600000
    const int* src = edge;                   // edge_index[0, :]
    const int* dst = edge + E;               // edge_index[1, :]

    // Workspace layout (floats): deg/deg_inv | h0 | aggr (reused) | h1
    float* deg  = (float*)d_ws;                       // N
    float* h0   = deg  + N;                           // N*128
    float* aggr = h0   + (size_t)N * SAGE_FEAT;       // N*128
    float* h1   = aggr + (size_t)N * SAGE_FEAT;       // N*128
    float* out  = (float*)d_out;                      // N*64

    const int mTiles = N / 16;               // 3125, exact

    // --- degree and inverse degree ---
    sage_zero_f32<<<cdiv(N, 256), 256, 0, stream>>>(deg, N);
    sage_deg_count<<<cdiv(E, 256), 256, 0, stream>>>(dst, deg, E);
    sage_deg_inv<<<cdiv(N, 256), 256, 0, stream>>>(deg, N);

    // --- h0 = x @ W_pre + b_pre  (50000 x 256 x 128), 1 strip/block ---
    sage_gemm_wmma<SAGE_IN, SAGE_FEAT, false><<<mTiles, 256, 0, stream>>>(
        x, W_pre, nullptr, nullptr, b_pre, nullptr, h0, mTiles, 0);

    // --- aggr1 = segment_sum(h0[src], dst) ---
    sage_zero_f32<<<cdiv(N * SAGE_FEAT, 256), 256, 0, stream>>>(aggr, N * SAGE_FEAT);
    sage_scatter_add<<<cdiv(E * (SAGE_FEAT / 4), 256), 256, 0, stream>>>(
        h0, src, dst, aggr, E, SAGE_FEAT);

    // --- h1 = relu((aggr1*deg_inv) @ Wl1 + h0 @ Wr1 + bl1), 1 strip/block ---
    sage_gemm_wmma<SAGE_FEAT, SAGE_HID, true><<<mTiles, 256, 0, stream>>>(
        aggr, Wl1, h0, Wr1, bl1, deg, h1, mTiles, 1);

    // --- aggr2 = segment_sum(h1[src], dst) ---
    sage_zero_f32<<<cdiv(N * SAGE_HID, 256), 256, 0, stream>>>(aggr, N * SAGE_HID);
    sage_scatter_add<<<cdiv(E * (SAGE_HID / 4), 256), 256, 0, stream>>>(
        h1, src, dst, aggr, E, SAGE_HID);

    // --- out = (aggr2*deg_inv) @ Wl2 + h1 @ Wr2 + bl2, 2 strips/block ---
    sage_gemm_wmma<SAGE_HID, SAGE_OUT, true><<<cdiv(mTiles, 2), 256, 0, stream>>>(
        aggr, Wl2, h1, Wr2, bl2, deg, out, mTiles, 0);

    // --- F.normalize(out, p=2, dim=-1) ---
    sage_l2norm_rows<<<cdiv(N * 32, 256), 256, 0, stream>>>(out, N);
    (void)n_in; (void)out_size; (void)ws_size;
}